// PrepareForDenseMotion_22479858827799
// MI455X (gfx1250) — compile-verified
//
#include <hip/hip_runtime.h>

typedef __attribute__((ext_vector_type(2))) float v2f;
typedef __attribute__((ext_vector_type(8))) float v8f;

#define HB   8
#define HK   10
#define HH   128
#define HW   128
#define HC   64
#define SIN  256          // source H_IN / W_IN
#define K1E  11           // K+1
#define CHN  67           // 64 warped + 1 heat + 2 kp
#define NCH  (K1E * CHN)  // 737

// -----------------------------------------------------------------------------
// Cross-lane sum of 64 fp32 partials (2 per lane) using V_WMMA_F32_16X16X4_F32.
// A = ones(16x4): lanes 0-15 supply A[m,0..1], lanes 16-31 supply A[m,2..3].
// B(4x16): vgpr0/vgpr1 of lane L hold B[0/1, L] (L<16) and B[2/3, L-16].
// D[m,n] = sum_k B[k,n]  =>  every lane's d[0] = colsum[lane % 16].
// Four xor-shuffle adds finish the 16-way sum; result broadcast to all lanes.
// EXEC is all-ones here (no divergence in this kernel before the WMMA).
// -----------------------------------------------------------------------------
__device__ __forceinline__ float wave_sum64_wmma(float p0, float p1) {
    v2f a; a[0] = 1.0f; a[1] = 1.0f;
    v2f b; b[0] = p0;   b[1] = p1;
    v8f c = {};
    c = __builtin_amdgcn_wmma_f32_16x16x4_f32(
            /*neg_a=*/false, a, /*neg_b=*/false, b,
            /*c_mod=*/(short)0, c, /*reuse_a=*/false, /*reuse_b=*/false);
    float s = c[0];
    s += __shfl_xor(s, 1, 32);
    s += __shfl_xor(s, 2, 32);
    s += __shfl_xor(s, 4, 32);
    s += __shfl_xor(s, 8, 32);
    return s;
}

// One block per (b,k): sums heatmap_driving[b,k,:,:] and heatmap_source[b,k,:,:].
__global__ void heat_sums_kernel(const float* __restrict__ hd,
                                 const float* __restrict__ hs,
                                 float* __restrict__ sums /* [B*K][2] */) {
    const int bk = blockIdx.x;        // 0..79
    const int t  = threadIdx.x;       // 0..255 (8 waves)
    const float* pd = hd + (size_t)bk * (HH * HW);
    const float* ps = hs + (size_t)bk * (HH * HW);

    float d0 = 0.f, d1 = 0.f, s0 = 0.f, s1 = 0.f;
    #pragma unroll 4
    for (int i = t; i < HH * HW; i += 512)       { d0 += pd[i]; s0 += ps[i]; }
    #pragma unroll 4
    for (int i = t + 256; i < HH * HW; i += 512) { d1 += pd[i]; s1 += ps[i]; }

    float dsum = wave_sum64_wmma(d0, d1);
    float ssum = wave_sum64_wmma(s0, s1);

    __shared__ float red[2][8];
    const int wave = t >> 5, lane = t & 31;
    if (lane == 0) { red[0][wave] = dsum; red[1][wave] = ssum; }
    __syncthreads();
    if (t == 0) {
        float a = 0.f, b = 0.f;
        #pragma unroll
        for (int w = 0; w < 8; ++w) { a += red[0][w]; b += red[1][w]; }
        sums[bk * 2 + 0] = a;
        sums[bk * 2 + 1] = b;
    }
}

// Main output: inputs (B, 737, 128, 128). One float per thread, writes fully
// coalesced (consecutive lanes = consecutive x). j and k are wave-uniform.
__global__ void dense_motion_kernel(const float* __restrict__ src,   // (8,64,256,256)
                                    const float* __restrict__ hd,    // (8,10,128,128)
                                    const float* __restrict__ hs,    // (8,10,128,128)
                                    const float* __restrict__ kpd,   // (8,10,2)
                                    const float* __restrict__ kps,   // (8,10,2)
                                    const float* __restrict__ sums,  // [80][2]
                                    float* __restrict__ out) {
    const unsigned i  = blockIdx.x * blockDim.x + threadIdx.x;
    const unsigned N1 = (unsigned)HB * NCH * HH * HW;
    if (i >= N1) return;

    const unsigned x    = i & (HW - 1);
    const unsigned y    = (i >> 7) & (HH - 1);
    const unsigned chb  = i >> 14;           // b*737 + ch
    const unsigned ch   = chb % NCH;
    const unsigned b    = chb / NCH;
    const unsigned k    = ch / CHN;          // 0..10
    const unsigned j    = ch - k * CHN;      // 0..66

    float dx = 0.f, dy = 0.f;
    if (k < HK) {
        const unsigned kpo = (b * HK + k) * 2;
        dx = kpd[kpo]     - kps[kpo];
        dy = kpd[kpo + 1] - kps[kpo + 1];
    }

    float val;
    if (j < HC) {
        // Undo the flat reshape: (y,x,j) -> warped (c, yy, xx)
        const unsigned c  = y >> 1;
        const unsigned yy = ((y & 1u) << 6) | (x >> 1);
        const unsigned xx = ((x & 1u) << 6) | j;

        // grid_sample_border of the nearest-resized source, uniform shift (dx,dy)
        float gx = 2.0f * ((float)xx * (1.0f / 127.0f)) - 1.0f + dx;
        float gy = 2.0f * ((float)yy * (1.0f / 127.0f)) - 1.0f + dy;
        float xs = ((gx + 1.0f) * 128.0f - 1.0f) * 0.5f;
        float ys = ((gy + 1.0f) * 128.0f - 1.0f) * 0.5f;
        xs = fminf(fmaxf(xs, 0.0f), 127.0f);
        ys = fminf(fmaxf(ys, 0.0f), 127.0f);
        const float x0f = floorf(xs), y0f = floorf(ys);
        const float wx = xs - x0f,   wy = ys - y0f;
        const int x0 = (int)x0f,     y0 = (int)y0f;
        const int x1 = min(x0 + 1, 127), y1 = min(y0 + 1, 127);

        // nearest-resize => sample source at doubled indices
        const float* plane = src + (size_t)(b * HC + c) * (SIN * SIN);
        const float v00 = plane[(2 * y0) * SIN + 2 * x0];
        const float v01 = plane[(2 * y0) * SIN + 2 * x1];
        const float v10 = plane[(2 * y1) * SIN + 2 * x0];
        const float v11 = plane[(2 * y1) * SIN + 2 * x1];
        val = v00 * (1.f - wy) * (1.f - wx) + v01 * (1.f - wy) * wx
            + v10 * wy * (1.f - wx)         + v11 * wy * wx;
    } else if (j == HC) {
        if (k < HK) {
            const unsigned ho = ((b * HK + k) << 14) | (y << 7) | x;
            const unsigned so = (b * HK + k) * 2;
            val = hd[ho] / sums[so] - hs[ho] / sums[so + 1];
        } else {
            val = 0.f;
        }
    } else {
        val = (j == HC + 1) ? dx : dy;   // already 0 for k == 10
    }
    out[i] = val;
}

// Second output: kp_ext (B, 11, 128, 128, 2), flat after the first output.
__global__ void kp_ext_kernel(const float* __restrict__ kpd,
                              const float* __restrict__ kps,
                              float* __restrict__ out2) {
    const unsigned i  = blockIdx.x * blockDim.x + threadIdx.x;
    const unsigned N2 = (unsigned)HB * K1E * HH * HW * 2;
    if (i >= N2) return;
    const unsigned comp = i & 1u;
    const unsigned bk   = i >> 15;       // (i>>1) >> 14  = b*11 + k
    const unsigned k    = bk % K1E;
    const unsigned b    = bk / K1E;
    float v = 0.f;
    if (k < HK) {
        const unsigned kpo = (b * HK + k) * 2 + comp;
        v = kpd[kpo] - kps[kpo];
    }
    out2[i] = v;
}

extern "C" void kernel_launch(void* const* d_in, const int* in_sizes, int n_in,
                              void* d_out, int out_size, void* d_ws, size_t ws_size,
                              hipStream_t stream) {
    const float* src = (const float*)d_in[0];  // source_image
    const float* hd  = (const float*)d_in[1];  // heatmap_driving
    const float* hs  = (const float*)d_in[2];  // heatmap_source
    const float* kpd = (const float*)d_in[3];  // kp_mean_driving
    const float* kps = (const float*)d_in[4];  // kp_mean_source
    float* out  = (float*)d_out;
    float* sums = (float*)d_ws;                // 80*2 floats of scratch

    heat_sums_kernel<<<HB * HK, 256, 0, stream>>>(hd, hs, sums);

    const unsigned N1 = (unsigned)HB * NCH * HH * HW;       // 96,600,064
    dense_motion_kernel<<<(N1 + 255) / 256, 256, 0, stream>>>(
        src, hd, hs, kpd, kps, sums, out);

    const unsigned N2 = (unsigned)HB * K1E * HH * HW * 2;   // 2,883,584
    kp_ext_kernel<<<(N2 + 255) / 256, 256, 0, stream>>>(kpd, kps, out + N1);
}